// SqueezeAndExcite_11416023073355
// MI455X (gfx1250) — compile-verified
//
#include <hip/hip_runtime.h>
#include <hip/hip_bf16.h>
#include <math.h>

#define C_CH   256
#define CS_CH  64
#define HW     12544   // 112*112
#define BATCH  32
#define THREADS 256

typedef __attribute__((ext_vector_type(2))) float v2f;
typedef __attribute__((ext_vector_type(4))) float v4f;
typedef __attribute__((ext_vector_type(8))) float v8f;

// ---------------------------------------------------------------------------
// Kernel 1: global average pool over H,W. One block per (b,c). 256 threads.
// Pure streaming read of x (411 MB) -> y[32*256].
// ---------------------------------------------------------------------------
__global__ void __launch_bounds__(THREADS)
se_pool_kernel(const float* __restrict__ x, float* __restrict__ y) {
    const int bc = blockIdx.x;                       // 0 .. 8191
    const v4f* xp = reinterpret_cast<const v4f*>(x + (size_t)bc * HW);
    const int n4 = HW / 4;                           // 3136 float4s per slice
    float s = 0.f;
    for (int i = threadIdx.x; i < n4; i += THREADS) {
        v4f v = xp[i];                               // global_load_b128
        s += v.x + v.y + v.z + v.w;
    }
    __shared__ float red[THREADS];
    red[threadIdx.x] = s;
    __syncthreads();
    #pragma unroll
    for (int off = THREADS / 2; off > 0; off >>= 1) {
        if (threadIdx.x < off) red[threadIdx.x] += red[threadIdx.x + off];
        __syncthreads();
    }
    if (threadIdx.x == 0) y[bc] = red[0] * (1.0f / (float)HW);
}

// ---------------------------------------------------------------------------
// Kernel 2: SE bottleneck via V_WMMA_F32_16X16X4_F32.
//   z    = swish(y @ w_reduce^T + b_reduce)      (32x64)
//   gate = sigmoid(z @ w_expand^T + b_expand)    (32x256)
// Single block, 256 threads = 8 wave32s. EXEC all-1s throughout (uniform flow).
//
// WMMA f32 16x16x4 per-lane layout (ISA 7.12.2):
//   A (16x4):  lane = kh*16 + m (m=lane&15, kh=lane>>4); a[0]=A[m][k0+2kh], a[1]=A[m][k0+2kh+1]
//   B (4x16):  lane = kh*16 + n;                         b[0]=B[k0+2kh][n], b[1]=B[k0+2kh+1][n]
//   C/D 16x16: VGPR r holds D[r + 8*kh][lane&15]
// ---------------------------------------------------------------------------
__global__ void __launch_bounds__(THREADS)
se_gate_kernel(const float* __restrict__ y,     // [32][256]
               const float* __restrict__ wr,    // [64][256]  (w_reduce[s][c])
               const float* __restrict__ br,    // [64]
               const float* __restrict__ we,    // [256][64]  (w_expand[c][s])
               const float* __restrict__ be,    // [256]
               float*       __restrict__ gate)  // [32][256]
{
    __shared__ float z[BATCH][CS_CH];            // 32x64 f32 = 8 KB
    const int lane = threadIdx.x & 31;
    const int wave = threadIdx.x >> 5;           // 0..7
    const int n    = lane & 15;
    const int kh   = lane >> 4;

    // ---- matmul 1: 2 (M) x 4 (N) tiles of 16x16, one tile per wave, K=256 in steps of 4
    {
        const int tm = wave >> 2;                // 0..1
        const int tn = wave & 3;                 // 0..3
        const int am = tm * 16 + n;              // A row for this lane
        const int sg = tn * 16 + n;              // B column (squeeze channel) for this lane
        v8f acc = {};
        for (int k0 = 0; k0 < C_CH; k0 += 4) {
            const int k = k0 + 2 * kh;
            v2f a, b;
            a.x = y[am * C_CH + k];
            a.y = y[am * C_CH + k + 1];
            b.x = wr[sg * C_CH + k];             // B[k][s] = w_reduce[s][k]
            b.y = wr[sg * C_CH + k + 1];
            acc = __builtin_amdgcn_wmma_f32_16x16x4_f32(
                false, a, false, b, (short)0, acc, false, false);
        }
        const float bias = br[sg];
        #pragma unroll
        for (int r = 0; r < 8; ++r) {
            const int mg = tm * 16 + r + 8 * kh;
            float v = acc[r] + bias;
            v = v / (1.0f + __expf(-v));         // swish
            z[mg][sg] = v;
        }
    }
    __syncthreads();

    // ---- matmul 2: 2 (M) x 16 (N) = 32 tiles, 4 per wave, K=64 in steps of 4
    for (int t = wave; t < 32; t += 8) {         // uniform per-wave trip count
        const int tm = t >> 4;                   // 0..1
        const int tn = t & 15;                   // 0..15
        const int am = tm * 16 + n;
        const int cg = tn * 16 + n;              // output channel for this lane
        v8f acc = {};
        for (int k0 = 0; k0 < CS_CH; k0 += 4) {
            const int k = k0 + 2 * kh;
            v2f a, b;
            a.x = z[am][k];
            a.y = z[am][k + 1];
            b.x = we[cg * CS_CH + k];            // B[k][c] = w_expand[c][k]
            b.y = we[cg * CS_CH + k + 1];
            acc = __builtin_amdgcn_wmma_f32_16x16x4_f32(
                false, a, false, b, (short)0, acc, false, false);
        }
        const float bias = be[cg];
        #pragma unroll
        for (int r = 0; r < 8; ++r) {
            const int mg = tm * 16 + r + 8 * kh;
            const float v = acc[r] + bias;
            gate[mg * C_CH + cg] = 1.0f / (1.0f + __expf(-v));
        }
    }
}

// ---------------------------------------------------------------------------
// Kernel 3: out = x * gate[b,c]. One block per (b,c); gate is block-uniform.
// Non-temporal hints: x is on its last use, out is write-once streaming.
// ---------------------------------------------------------------------------
__global__ void __launch_bounds__(THREADS)
se_scale_kernel(const float* __restrict__ x,
                const float* __restrict__ gate,
                float*       __restrict__ out) {
    const int bc = blockIdx.x;
    const float g = gate[bc];                    // uniform -> scalar load
    const v4f* xp = reinterpret_cast<const v4f*>(x + (size_t)bc * HW);
    v4f*       op = reinterpret_cast<v4f*>(out + (size_t)bc * HW);
    const int n4 = HW / 4;
    for (int i = threadIdx.x; i < n4; i += THREADS) {
        v4f v = __builtin_nontemporal_load(&xp[i]);
        v.x *= g; v.y *= g; v.z *= g; v.w *= g;
        __builtin_nontemporal_store(v, &op[i]);
    }
}

// ---------------------------------------------------------------------------
extern "C" void kernel_launch(void* const* d_in, const int* in_sizes, int n_in,
                              void* d_out, int out_size, void* d_ws, size_t ws_size,
                              hipStream_t stream) {
    const float* x  = (const float*)d_in[0];   // [32,256,112,112]
    const float* wr = (const float*)d_in[1];   // [64,256]
    const float* br = (const float*)d_in[2];   // [64]
    const float* we = (const float*)d_in[3];   // [256,64]
    const float* be = (const float*)d_in[4];   // [256]
    float* out = (float*)d_out;

    float* y    = (float*)d_ws;                // 32*256 floats
    float* gate = y + BATCH * C_CH;            // 32*256 floats

    const int n_bc = BATCH * C_CH;             // 8192
    se_pool_kernel <<<n_bc, THREADS, 0, stream>>>(x, y);
    se_gate_kernel <<<1,    THREADS, 0, stream>>>(y, wr, br, we, be, gate);
    se_scale_kernel<<<n_bc, THREADS, 0, stream>>>(x, gate, out);
}